// GaussianVectorQuantizer_62586263437871
// MI455X (gfx1250) — compile-verified
//
#include <hip/hip_runtime.h>
#include <hip/hip_bf16.h>
#include <stdint.h>

typedef float v2f __attribute__((ext_vector_type(2)));
typedef float v8f __attribute__((ext_vector_type(8)));

#define K_CODES   1024
#define DIMZ      64
#define N_ROWS    65536
#define N_TILES   4096     // N_ROWS / 16
#define CB_STRIDE 68       // padded row stride (floats): stride%64==4 -> conflict-free A reads
#define LN2F      0.69314718055994530942f
#define LOG2EF    1.4426950408889634074f

// ws layout (floats / uints):
//  [0    .. 1023]  hist (uint32)
//  [1024 .. 5119]  per-tile sum(p*log p)           (deterministic partials)
//  [5120 .. 9215]  per-tile w*||z||^2 - s_max      (deterministic partials)
//  [9216 ..10239]  -0.5*||c_k||^2
#define WS_HIST 0
#define WS_KD   1024
#define WS_KC   5120
#define WS_CN   9216

__device__ __forceinline__ float fexp2(float x) { return __builtin_amdgcn_exp2f(x); }
__device__ __forceinline__ float flog2(float x) { return __builtin_amdgcn_logf(x); }

// ---------------- kernel 0: zero hist, precompute -0.5*||c||^2 ----------------
__global__ void vq_prep(const float* __restrict__ codebook, float* __restrict__ wsf) {
    int tid = blockIdx.x * blockDim.x + threadIdx.x;
    if (tid < K_CODES) {
        reinterpret_cast<unsigned*>(wsf)[WS_HIST + tid] = 0u;
        const float4* row = reinterpret_cast<const float4*>(codebook + tid * DIMZ);
        float s = 0.0f;
        #pragma unroll
        for (int i = 0; i < DIMZ / 4; ++i) {
            float4 v = row[i];
            s = fmaf(v.x, v.x, fmaf(v.y, v.y, fmaf(v.z, v.z, fmaf(v.w, v.w, s))));
        }
        wsf[WS_CN + tid] = -0.5f * s;
    }
}

// ---------------- kernel 1: main fused VQ ----------------
__global__ __launch_bounds__(512, 1)
void vq_main(const float* __restrict__ z, const float* __restrict__ var_q,
             const float* __restrict__ codebook, float* __restrict__ out,
             float* __restrict__ wsf) {
    // 286720 B of LDS: codebook (padded) + -0.5||c||^2 + histogram
    __shared__ __align__(16) float    cbS[K_CODES * CB_STRIDE];
    __shared__ __align__(16) float    cnS[K_CODES];
    __shared__            unsigned    histS[K_CODES];

    const int tid = threadIdx.x;

    // cooperative fill: whole codebook into LDS, padded to CB_STRIDE
    for (int i = tid; i < K_CODES * (DIMZ / 4); i += blockDim.x) {
        int k  = i >> 4;
        int d4 = (i & 15) << 2;
        float4 v = *reinterpret_cast<const float4*>(codebook + k * DIMZ + d4);
        *reinterpret_cast<float4*>(&cbS[k * CB_STRIDE + d4]) = v;
    }
    for (int i = tid; i < K_CODES; i += blockDim.x) {
        cnS[i]   = wsf[WS_CN + i];
        histS[i] = 0u;
    }
    __syncthreads();

    const float w      = 0.5f / fmaxf(var_q[0], 1e-10f);
    const float uScale = 2.0f * w * LOG2EF;   // logits kept in base-2 domain

    const int lane = threadIdx.x & 31;
    const int h    = lane >> 4;               // half of wave
    const int m_   = lane & 15;
    const int wid  = (blockIdx.x * blockDim.x + threadIdx.x) >> 5;
    const int nw   = (gridDim.x * blockDim.x) >> 5;

    for (int tile = wid; tile < N_TILES; tile += nw) {
        // ---- load 16 hoisted B fragments (z rows) + partial ||z||^2 ----
        const int n  = tile * 16 + m_;
        const int b_ = n >> 10;
        const int wh = n & 1023;
        const float* zrow = z + (b_ * 65536 + wh);   // element c at zrow[c*1024]
        v2f  bf[16];
        float znp = 0.0f;
        #pragma unroll
        for (int kk = 0; kk < 16; ++kk) {
            const int d0 = 4 * kk + 2 * h;
            float x = zrow[d0 * 1024];
            float y = zrow[d0 * 1024 + 1024];
            bf[kk].x = x; bf[kk].y = y;
            znp = fmaf(x, x, fmaf(y, y, znp));
        }

        // per-slot online softmax state (8 code lanes per thread)
        float mu[8], Zs[8], Ss[8];
        int   ix[8];
        #pragma unroll
        for (int r = 0; r < 8; ++r) { mu[r] = -1e30f; Zs[r] = 0.f; Ss[r] = 0.f; ix[r] = 0; }

        for (int t = 0; t < 64; ++t) {
            // C operand = -0.5*||c||^2 for this slot's codes
            const float4* cn = reinterpret_cast<const float4*>(&cnS[t * 16 + 8 * h]);
            float4 c0 = cn[0], c1 = cn[1];
            v8f acc;
            acc[0] = c0.x; acc[1] = c0.y; acc[2] = c0.z; acc[3] = c0.w;
            acc[4] = c1.x; acc[5] = c1.y; acc[6] = c1.z; acc[7] = c1.w;

            const int rowoff = (t * 16 + m_) * CB_STRIDE + 2 * h;
            #pragma unroll
            for (int kk = 0; kk < 16; ++kk) {
                v2f a = *reinterpret_cast<const v2f*>(&cbS[rowoff + 4 * kk]);
                acc = __builtin_amdgcn_wmma_f32_16x16x4_f32(
                        false, a, false, bf[kk], (short)0, acc, false, false);
            }

            const int code0 = t * 16 + 8 * h;
            #pragma unroll
            for (int r = 0; r < 8; ++r) {
                float v  = acc[r] * uScale;          // logit in log2 units
                float mo = mu[r];
                float mn = fmaxf(mo, v);
                float ea = fexp2(mo - mn);
                float ev = fexp2(v  - mn);
                Ss[r] = (Ss[r] + Zs[r] * (mo - mn)) * ea + (v - mn) * ev;
                Zs[r] = Zs[r] * ea + ev;
                ix[r] = (v > mo) ? (code0 + r) : ix[r];
                mu[r] = mn;
            }
        }

        // ---- fold 8 slots within lane ----
        float M = mu[0], Z = Zs[0], S = Ss[0];
        int   I = ix[0];
        #pragma unroll
        for (int r = 1; r < 8; ++r) {
            float mn = fmaxf(M, mu[r]);
            float ea = fexp2(M - mn), eb = fexp2(mu[r] - mn);
            S = (S + Z * (M - mn)) * ea + (Ss[r] + Zs[r] * (mu[r] - mn)) * eb;
            Z = Z * ea + Zs[r] * eb;
            I = (mu[r] > M) ? ix[r] : I;
            M = mn;
        }
        // ---- cross-half combine (codes split by lane half) ----
        {
            float mB = __shfl_xor(M, 16);
            float ZB = __shfl_xor(Z, 16);
            float SB = __shfl_xor(S, 16);
            int   IB = __shfl_xor(I, 16);
            float mn = fmaxf(M, mB);
            float ea = fexp2(M - mn), eb = fexp2(mB - mn);
            S = (S + Z * (M - mn)) * ea + (SB + ZB * (mB - mn)) * eb;
            Z = Z * ea + ZB * eb;
            // symmetric tie-break so both halves agree (prefer smaller index)
            I = (mB > M || (mB == M && IB < I)) ? IB : I;
            M = mn;
        }
        float zn = znp + __shfl_xor(znp, 16);    // full ||z_row||^2

        // per-row loss terms
        float rowKd = LN2F * (S / Z - flog2(Z));       // sum_k p*log p for this row
        float rowKc = fmaf(w, zn, -M * LN2F);          // w*||z||^2 - s_max = w*d2_argmax
        #pragma unroll
        for (int off = 1; off < 16; off <<= 1) {
            rowKd += __shfl_xor(rowKd, off);
            rowKc += __shfl_xor(rowKc, off);
        }
        if (lane == 0) {
            wsf[WS_KD + tile] = rowKd;
            wsf[WS_KC + tile] = rowKc;
        }
        if (lane < 16) atomicAdd(&histS[I], 1u);

        // ---- scatter z_q = codebook[argmax] back to [bs, dim, W, H] layout ----
        {
            float* obase = out + (b_ * 65536 + wh);
            const float* crow = &cbS[I * CB_STRIDE + 32 * h];
            #pragma unroll
            for (int c = 0; c < 32; ++c)
                obase[(c + 32 * h) * 1024] = crow[c];   // coalesced 64B per step per half
        }
    }

    __syncthreads();
    for (int i = tid; i < K_CODES; i += blockDim.x) {
        unsigned v = histS[i];
        if (v) atomicAdd(&reinterpret_cast<unsigned*>(wsf)[WS_HIST + i], v);
    }
}

// ---------------- kernel 2: finalize loss + perplexity ----------------
__global__ void vq_final(const float* __restrict__ wsf, float* __restrict__ out) {
    __shared__ float red0[256], red1[256], red2[256];
    const int tid = threadIdx.x;
    float kd = 0.f, kc = 0.f, pe = 0.f;
    for (int t = tid; t < N_TILES; t += 256) {
        kd += wsf[WS_KD + t];
        kc += wsf[WS_KC + t];
    }
    const unsigned* hist = reinterpret_cast<const unsigned*>(wsf) + WS_HIST;
    for (int k = tid; k < K_CODES; k += 256) {
        float avg = (float)hist[k] * (1.0f / (float)N_ROWS);
        pe += avg * (flog2(avg + 1e-7f) * LN2F);    // avg * ln(avg + eps)
    }
    red0[tid] = kd; red1[tid] = kc; red2[tid] = pe;
    __syncthreads();
    for (int s = 128; s > 0; s >>= 1) {
        if (tid < s) { red0[tid] += red0[tid + s]; red1[tid] += red1[tid + s]; red2[tid] += red2[tid + s]; }
        __syncthreads();
    }
    if (tid == 0) {
        out[4194304] = (red0[0] + red1[0]) * (1.0f / 64.0f);       // loss
        out[4194305] = fexp2(-red2[0] * LOG2EF);                    // perplexity
    }
}

extern "C" void kernel_launch(void* const* d_in, const int* in_sizes, int n_in,
                              void* d_out, int out_size, void* d_ws, size_t ws_size,
                              hipStream_t stream) {
    const float* z        = (const float*)d_in[0];   // [64,64,32,32]
    const float* var_q    = (const float*)d_in[1];   // [1]
    const float* codebook = (const float*)d_in[2];   // [1024,64]
    float* out = (float*)d_out;                      // 4194304 z_q + loss + perplexity
    float* wsf = (float*)d_ws;                       // needs 40 KB

    vq_prep <<<4, 256, 0, stream>>>(codebook, wsf);
    vq_main <<<256, 512, 0, stream>>>(z, var_q, codebook, out, wsf);
    vq_final<<<1, 256, 0, stream>>>(wsf, out);
}